// GCN_23673859736445
// MI455X (gfx1250) — compile-verified
//
#include <hip/hip_runtime.h>

#define N_NODES 50000
#define F 128

typedef float v2f __attribute__((ext_vector_type(2)));
typedef float v8f __attribute__((ext_vector_type(8)));

__device__ __forceinline__ void atomAddF32(float* p, float v) {
    __hip_atomic_fetch_add(p, v, __ATOMIC_RELAXED, __HIP_MEMORY_SCOPE_AGENT);
}

// ---------------------------------------------------------------------------
// Degree / norm kernels
// ---------------------------------------------------------------------------
__global__ __launch_bounds__(256) void k_zero(float* __restrict__ p, int n) {
    int i = blockIdx.x * 256 + threadIdx.x;
    if (i < n) p[i] = 0.0f;
}

__global__ __launch_bounds__(256) void k_degree(const long long* __restrict__ dst,
                                                float* __restrict__ deg, long long E) {
    long long e = (long long)blockIdx.x * 256 + threadIdx.x;
    if (e < E) atomAddF32(&deg[dst[e]], 1.0f);
}

__global__ __launch_bounds__(256) void k_dinv(float* __restrict__ deg_dinv, int n) {
    int i = blockIdx.x * 256 + threadIdx.x;
    if (i < n) deg_dinv[i] = rsqrtf(1.0f + deg_dinv[i]);   // in-place deg -> d^-1/2
}

// ---------------------------------------------------------------------------
// Fused WMMA fp32 GEMM + self-loop/bias:
//   h[n]   = act(in[n]) @ W
//   agg[n] = dinv[n]^2 * h[n] + b
// Wave computes 32 rows x 128 cols via V_WMMA_F32_16X16X4_F32.
// W staged into LDS pair-interleaved: sW2[k/2][c] = {W[k][c], W[k+1][c]}
// so each B fragment is a single aligned ds_load_b64.
// A layout: lanes 0-15 -> rows M=lane, K=kb+{0,1}; lanes 16-31 -> K=kb+{2,3}.
// C/D: vgpr v -> row = base + v + 8*(lane>>4), col = lane&15.
// Safe when in == agg: each A-row is read only by the wave that writes it
// (fully-OOB waves exit before any read; partial waves clamp into own range).
// ---------------------------------------------------------------------------
template <bool RELU>
__global__ __launch_bounds__(256) void k_gemm128(const float* __restrict__ in,
                                                 const float* __restrict__ W,
                                                 const float* __restrict__ b,
                                                 const float* __restrict__ dinv,
                                                 float* __restrict__ h,
                                                 float* __restrict__ agg, int nrows) {
    __shared__ v2f sW2[64 * F];  // 64 KB pair-interleaved (WGP has 320 KB LDS)
    const int tid = threadIdx.x;
    for (int j = tid; j < 64 * 64; j += 256) {
        int p = j >> 6;             // K-pair index 0..63
        int c = (j & 63) * 2;       // even column
        float2 r0 = *(const float2*)&W[(2 * p) * F + c];
        float2 r1 = *(const float2*)&W[(2 * p + 1) * F + c];
        float4 st; st.x = r0.x; st.y = r1.x; st.z = r0.y; st.w = r1.y;
        *(float4*)&sW2[p * F + c] = st;   // 16B aligned (c even)
    }
    __syncthreads();

    const int wave = tid >> 5;
    const int lane = tid & 31;
    const int half = lane >> 4;
    const int l16  = lane & 15;
    const long long row0 = (long long)blockIdx.x * 256 + wave * 32;
    if (row0 >= nrows) return;                 // wave-uniform exit (EXEC stays full)

    long long arow0 = row0 + l16;
    long long arow1 = row0 + 16 + l16;
    if (arow0 > nrows - 1) arow0 = nrows - 1;  // clamps stay inside this wave's rows
    if (arow1 > nrows - 1) arow1 = nrows - 1;
    const float* ap0 = in + arow0 * F + half * 2;
    const float* ap1 = in + arow1 * F + half * 2;

    v8f acc0[8], acc1[8];
    v8f zero = {};
#pragma unroll
    for (int t = 0; t < 8; ++t) { acc0[t] = zero; acc1[t] = zero; }

#pragma unroll 2
    for (int kb = 0; kb < F; kb += 4) {
        float2 av0 = *(const float2*)(ap0 + kb);
        float2 av1 = *(const float2*)(ap1 + kb);
        if (RELU) {
            av0.x = fmaxf(av0.x, 0.f); av0.y = fmaxf(av0.y, 0.f);
            av1.x = fmaxf(av1.x, 0.f); av1.y = fmaxf(av1.y, 0.f);
        }
        v2f a0; a0.x = av0.x; a0.y = av0.y;
        v2f a1; a1.x = av1.x; a1.y = av1.y;
        const int p = (kb >> 1) + half;        // pair-row in LDS
#pragma unroll
        for (int t = 0; t < 8; ++t) {
            v2f bfr = sW2[p * F + t * 16 + l16];   // single ds_load_b64
            acc0[t] = __builtin_amdgcn_wmma_f32_16x16x4_f32(
                false, a0, false, bfr, (short)0, acc0[t], false, false);
            acc1[t] = __builtin_amdgcn_wmma_f32_16x16x4_f32(
                false, a1, false, bfr, (short)0, acc1[t], false, false);
        }
    }

    float bias[8];
#pragma unroll
    for (int t = 0; t < 8; ++t) bias[t] = b[t * 16 + l16];

#pragma unroll
    for (int rt = 0; rt < 2; ++rt) {
#pragma unroll
        for (int v = 0; v < 8; ++v) {
            long long r = row0 + rt * 16 + v + 8 * half;
            if (r < nrows) {
                float dv = dinv[r];
                float ns = dv * dv;
#pragma unroll
                for (int t = 0; t < 8; ++t) {
                    float hv = rt == 0 ? acc0[t][v] : acc1[t][v];
                    long long idx = r * F + t * 16 + l16;
                    h[idx]   = hv;
                    agg[idx] = ns * hv + bias[t];
                }
            }
        }
    }
}

// ---------------------------------------------------------------------------
// Edge scatter: agg[dst] += dinv[src]*dinv[dst] * h[src]
// one wave per edge; lane handles 4 features (float4 gather, 4 f32 atomics)
// ---------------------------------------------------------------------------
__global__ __launch_bounds__(256) void k_scatter(const long long* __restrict__ src,
                                                 const long long* __restrict__ dst,
                                                 const float* __restrict__ dinv,
                                                 const float* __restrict__ h,
                                                 float* __restrict__ agg, long long E) {
    long long e = (long long)blockIdx.x * 8 + (threadIdx.x >> 5);
    if (e >= E) return;
    const int f = (threadIdx.x & 31) * 4;
    long long s = src[e], d = dst[e];
    float w = dinv[s] * dinv[d];
    float4 hv = *(const float4*)&h[s * F + f];
    float* a = &agg[d * F + f];
    atomAddF32(a + 0, w * hv.x);
    atomAddF32(a + 1, w * hv.y);
    atomAddF32(a + 2, w * hv.z);
    atomAddF32(a + 3, w * hv.w);
}

// ---------------------------------------------------------------------------
// Final layer (F_out = 1), fused:
//   h4[n]  = dot(relu(agg[n]), W4)
//   out[n] = b4 + dinv[n]^2 * h4[n]        (self-loop + bias init)
// ---------------------------------------------------------------------------
__global__ __launch_bounds__(256) void k_final_dot(const float* __restrict__ agg,
                                                   const float* __restrict__ W4,
                                                   const float* __restrict__ b4,
                                                   const float* __restrict__ dinv,
                                                   float* __restrict__ h4,
                                                   float* __restrict__ out, int n) {
    int node = blockIdx.x * 8 + (threadIdx.x >> 5);
    int lane = threadIdx.x & 31;
    if (node >= n) return;
    float4 a = *(const float4*)&agg[(long long)node * F + lane * 4];
    float4 w = *(const float4*)&W4[lane * 4];
    float s = fmaxf(a.x, 0.f) * w.x + fmaxf(a.y, 0.f) * w.y +
              fmaxf(a.z, 0.f) * w.z + fmaxf(a.w, 0.f) * w.w;
#pragma unroll
    for (int off = 16; off > 0; off >>= 1) s += __shfl_xor(s, off, 32);
    if (lane == 0) {
        h4[node] = s;
        float dv = dinv[node];
        out[node] = b4[0] + dv * dv * s;
    }
}

__global__ __launch_bounds__(256) void k_final_scatter(const long long* __restrict__ src,
                                                       const long long* __restrict__ dst,
                                                       const float* __restrict__ dinv,
                                                       const float* __restrict__ h4,
                                                       float* __restrict__ out, long long E) {
    long long e = (long long)blockIdx.x * 256 + threadIdx.x;
    if (e >= E) return;
    long long s = src[e], d = dst[e];
    atomAddF32(&out[d], dinv[s] * dinv[d] * h4[s]);
}

// ---------------------------------------------------------------------------
extern "C" void kernel_launch(void* const* d_in, const int* in_sizes, int n_in,
                              void* d_out, int out_size, void* d_ws, size_t ws_size,
                              hipStream_t stream) {
    const float* x  = (const float*)d_in[0];
    const long long* ei = (const long long*)d_in[1];
    const float* W1 = (const float*)d_in[2]; const float* b1 = (const float*)d_in[3];
    const float* W2 = (const float*)d_in[4]; const float* b2 = (const float*)d_in[5];
    const float* W3 = (const float*)d_in[6]; const float* b3 = (const float*)d_in[7];
    const float* W4 = (const float*)d_in[8]; const float* b4 = (const float*)d_in[9];

    const int N = N_NODES;
    const long long E = (long long)in_sizes[1] / 2;
    const long long* src = ei;
    const long long* dst = ei + E;

    // workspace carve (floats): dinv[Npad] | h[N*F] | agg[N*F] | h4[Npad]
    const long long Npad = (N + 255) & ~255LL;
    float* ws   = (float*)d_ws;
    float* dinv = ws;
    float* h    = dinv + Npad;
    float* agg  = h + (long long)N * F;
    float* h4   = agg + (long long)N * F;
    float* out  = (float*)d_out;

    const int nBlkN    = (N + 255) / 256;
    const int nBlkE    = (int)((E + 255) / 256);
    const int nBlkE8   = (int)((E + 7) / 8);
    const int nBlkGemm = (N + 255) / 256;      // 256 rows per block (8 waves x 32)
    const int nBlkN8   = (N + 7) / 8;

    // degrees -> dinv
    k_zero<<<nBlkN, 256, 0, stream>>>(dinv, N);
    k_degree<<<nBlkE, 256, 0, stream>>>(dst, dinv, E);
    k_dinv<<<nBlkN, 256, 0, stream>>>(dinv, N);

    // layer 1 (input x, no relu on input)
    k_gemm128<false><<<nBlkGemm, 256, 0, stream>>>(x, W1, b1, dinv, h, agg, N);
    k_scatter<<<nBlkE8, 256, 0, stream>>>(src, dst, dinv, h, agg, E);

    // layer 2 (relu fused into A-operand load; in-place agg read/write is safe)
    k_gemm128<true><<<nBlkGemm, 256, 0, stream>>>(agg, W2, b2, dinv, h, agg, N);
    k_scatter<<<nBlkE8, 256, 0, stream>>>(src, dst, dinv, h, agg, E);

    // layer 3
    k_gemm128<true><<<nBlkGemm, 256, 0, stream>>>(agg, W3, b3, dinv, h, agg, N);
    k_scatter<<<nBlkE8, 256, 0, stream>>>(src, dst, dinv, h, agg, E);

    // layer 4: 128 -> 1 (dot + self-loop/bias fused), then edge scatter
    k_final_dot<<<nBlkN8, 256, 0, stream>>>(agg, W4, b4, dinv, h4, out, N);
    k_final_scatter<<<nBlkE, 256, 0, stream>>>(src, dst, dinv, h4, out, E);
}